// ROIAlign_2267742732677
// MI455X (gfx1250) — compile-verified
//
#include <hip/hip_runtime.h>

typedef float v2f __attribute__((ext_vector_type(2)));
typedef float v4f __attribute__((ext_vector_type(4)));
typedef float v8f __attribute__((ext_vector_type(8)));

#define IN_H   100
#define IN_W   100
#define IN_C   256
#define OUT_HW 7
#define NSAMP  14   // OUT_HW * SAMPLING_RATIO
#define NTAP   28   // 2 taps per sample

__global__ __launch_bounds__(128)
void roi_align_wmma_kernel(const float* __restrict__ in,
                           const float* __restrict__ boxes,
                           const int*   __restrict__ box_inds,
                           float* __restrict__ out)
{
    // Per-wave x-interpolated tile: T[tap_row 0..27][ow 0..6][16 channels]
    __shared__ float Tl[4][NTAP * OUT_HW * 16];
    __shared__ float cx_s[NTAP], cy_s[NTAP];
    __shared__ int   Xcol_s[NTAP], Yrow_s[NTAP];

    const int box  = blockIdx.x;
    const int tid  = threadIdx.x;
    const int wave = tid >> 5;
    const int lane = tid & 31;

    // ---- setup: per-sample coords/weights (threads 0..27) ----
    if (tid < 2 * NSAMP) {
        const bool isY = (tid >= NSAMP);
        const int  k   = isY ? (tid - NSAMP) : tid;
        const int  dimMax = (isY ? IN_H : IN_W) - 1;
        const float lo = boxes[box * 4 + (isY ? 1 : 0)] * 0.125f;
        const float hi = boxes[box * 4 + (isY ? 3 : 2)] * 0.125f;
        const float sp = (hi - lo) * (1.0f / (float)NSAMP);
        const float s  = lo + sp * ((float)k + 0.5f) - 0.5f;
        const float fl = floorf(s);
        const float t  = s - fl;
        const int   ilo = min(max((int)fl, 0), dimMax);
        const int   ihi = min(max((int)fl + 1, 0), dimMax);
        const float v   = (s >= 0.0f && s <= (float)dimMax) ? 1.0f : 0.0f;
        const float w0  = 0.5f * v * (1.0f - t);   // 0.5 folds the 2x2 pool average
        const float w1  = 0.5f * v * t;
        if (isY) {
            Yrow_s[2*k] = ilo;  Yrow_s[2*k+1] = ihi;
            cy_s[2*k]   = w0;   cy_s[2*k+1]   = w1;
        } else {
            Xcol_s[2*k] = ilo;  Xcol_s[2*k+1] = ihi;
            cx_s[2*k]   = w0;   cx_s[2*k+1]   = w1;
        }
    }
    __syncthreads();

    const int bi = box_inds[box];
    const float* inb = in + (size_t)bi * (IN_H * IN_W * IN_C);
    float* Tw = Tl[wave];

    const int m     = lane & 15;          // A row / B col / D col
    const int khalf = (lane >> 4) << 1;   // k = khalf + component

    // ---- A-operand (y-combine weights), invariant across ct and ow ----
    // A[m, 4k0+k] = cy[4k0+k] when m==k0 else 0 (block structure of the
    // 7x28 pooling matrix). Built once: unconditional LDS loads + select.
    v2f Aop[7];
#pragma unroll
    for (int k0 = 0; k0 < 7; ++k0) {
        const float c0 = cy_s[4*k0 + khalf];
        const float c1 = cy_s[4*k0 + khalf + 1];
        const float hit = (m == k0) ? 1.0f : 0.0f;
        Aop[k0].x = hit * c0;
        Aop[k0].y = hit * c1;
    }

    // stage-1 lane mapping: 8 (taprow,ow) pairs x 4 channel-groups of float4
    const int sub = lane >> 2;          // 0..7
    const int cg  = (lane & 3) << 2;    // 0,4,8,12

    for (int ct = wave * 4; ct < wave * 4 + 4; ++ct) {
        const int cbase = ct * 16;

        // ---- stage 1: x-interpolation (gather, coalesced float4) ----
        for (int s0 = 0; s0 < NTAP * OUT_HW; s0 += 8) {
            const int idx = s0 + sub;
            if (idx < NTAP * OUT_HW) {
                const int i  = idx / OUT_HW;
                const int ow = idx - i * OUT_HW;
                const float* row = inb + (size_t)Yrow_s[i] * (IN_W * IN_C) + cbase + cg;
                v4f acc4 = {0.0f, 0.0f, 0.0f, 0.0f};
#pragma unroll
                for (int q = 0; q < 4; ++q) {
                    const float w  = cx_s[4*ow + q];
                    const v4f  val = *(const v4f*)(row + (size_t)Xcol_s[4*ow + q] * IN_C);
                    acc4 += w * val;
                }
                *(v4f*)(&Tw[(i * OUT_HW + ow) * 16 + cg]) = acc4;
            }
        }
        // T is wave-private: same-wave LDS ordering, no barrier needed.

        // ---- stage 2: y-combine as D = A x B on the matrix unit ----
        // V_WMMA_F32_16X16X4_F32, K=28 as 7 native K=4 steps.
        for (int ow = 0; ow < OUT_HW; ++ow) {
            v8f acc = {};
#pragma unroll
            for (int k0 = 0; k0 < 7; ++k0) {
                v2f b;
                const int i0 = 4*k0 + khalf;
                b.x = Tw[( i0      * OUT_HW + ow) * 16 + m];
                b.y = Tw[((i0 + 1) * OUT_HW + ow) * 16 + m];
                acc = __builtin_amdgcn_wmma_f32_16x16x4_f32(
                          /*neg_a=*/false, Aop[k0], /*neg_b=*/false, b,
                          /*c_mod=*/(short)0, acc,
                          /*reuse_a=*/false, /*reuse_b=*/false);
            }
            // D rows m=0..7 live in comps 0..7 of lanes 0..15 (n = lane)
            if (lane < 16) {
                float* op = out + (((size_t)box * OUT_HW) * OUT_HW + ow) * IN_C + cbase + lane;
#pragma unroll
                for (int r = 0; r < OUT_HW; ++r)
                    op[(size_t)r * (OUT_HW * IN_C)] = acc[r];
            }
        }
    }
}

extern "C" void kernel_launch(void* const* d_in, const int* in_sizes, int n_in,
                              void* d_out, int out_size, void* d_ws, size_t ws_size,
                              hipStream_t stream) {
    const float* in       = (const float*)d_in[0];   // (2,100,100,256) f32
    const float* boxes    = (const float*)d_in[1];   // (1024,4) f32
    const int*   box_inds = (const int*)d_in[2];     // (1024,) i32
    float* out = (float*)d_out;                      // (1024,7,7,256) f32
    (void)in_sizes; (void)n_in; (void)out_size; (void)d_ws; (void)ws_size;

    roi_align_wmma_kernel<<<1024, 128, 0, stream>>>(in, boxes, box_inds, out);
}